// _ProposalLayer_FPN_4440996184678
// MI455X (gfx1250) — compile-verified
//
#include <hip/hip_runtime.h>
#include <cstdint>
#include <cstddef>

#define NEG_SCORE     (-1.0e9f)
#define VALID_THRESH  (-0.5e9f)
#define NMS_THRESH_C  (0.7f)
#define TOPN          1000
#define NMS_BLOCK     1024

#if defined(__has_builtin)
#  if __has_builtin(__builtin_amdgcn_tensor_load_to_lds)
#    define HAVE_TDM 1
#  endif
#  if __has_builtin(__builtin_amdgcn_s_wait_tensorcnt)
#    define HAVE_TCNT 1
#  endif
#endif

#if defined(HAVE_TDM)
#  pragma message("CDNA5: tensor_load_to_lds builtin path ENABLED")
#else
#  pragma message("CDNA5: tensor_load_to_lds builtin NOT available -> LDS copy fallback")
#endif
#if defined(HAVE_TCNT)
#  pragma message("CDNA5: s_wait_tensorcnt builtin path ENABLED")
#endif

typedef unsigned int v4u __attribute__((ext_vector_type(4)));
typedef int          v8i __attribute__((ext_vector_type(8)));
typedef int          v4i __attribute__((ext_vector_type(4)));

// ---------------------------------------------------------------------------
// Kernel 1: init per-batch coordinate max (float bits; coords >= 0 so 0 ok)
// ---------------------------------------------------------------------------
__global__ void pl_init_kernel(int* __restrict__ maxbits, int B) {
    int i = blockIdx.x * blockDim.x + threadIdx.x;
    if (i < B) maxbits[i] = 0;  // 0.0f
}

// ---------------------------------------------------------------------------
// Kernel 2: bbox_transform_inv + clip_boxes, atomicMax of coords per batch.
// ---------------------------------------------------------------------------
__global__ void pl_decode_kernel(const float* __restrict__ deltas,
                                 const float* __restrict__ anchors,
                                 const float* __restrict__ im_info,
                                 float* __restrict__ proposals,
                                 int* __restrict__ maxbits,
                                 int B, int N) {
    const int total = B * N;
    const float4* a4 = reinterpret_cast<const float4*>(anchors);
    const float4* d4 = reinterpret_cast<const float4*>(deltas);
    float4* p4 = reinterpret_cast<float4*>(proposals);
    for (int id = blockIdx.x * blockDim.x + threadIdx.x; id < total;
         id += gridDim.x * blockDim.x) {
        const int b = id / N;
        const int j = id - b * N;
        const float4 a = a4[j];
        const float4 d = d4[id];
        const float ws = a.z - a.x + 1.0f;
        const float hs = a.w - a.y + 1.0f;
        const float cx = a.x + 0.5f * ws;
        const float cy = a.y + 0.5f * hs;
        const float pcx = d.x * ws + cx;
        const float pcy = d.y * hs + cy;
        const float pw = __expf(d.z) * ws;   // v_exp_f32 transcendental
        const float ph = __expf(d.w) * hs;
        const float wmax = im_info[b * 3 + 1] - 1.0f;
        const float hmax = im_info[b * 3 + 0] - 1.0f;
        const float x1 = fminf(fmaxf(pcx - 0.5f * pw, 0.0f), wmax);
        const float y1 = fminf(fmaxf(pcy - 0.5f * ph, 0.0f), hmax);
        const float x2 = fminf(fmaxf(pcx + 0.5f * pw, 0.0f), wmax);
        const float y2 = fminf(fmaxf(pcy + 0.5f * ph, 0.0f), hmax);
        p4[id] = make_float4(x1, y1, x2, y2);
        const float m = fmaxf(fmaxf(x1, y1), fmaxf(x2, y2));
        atomicMax(&maxbits[b], __float_as_int(m));  // valid: m >= 0
    }
}

// ---------------------------------------------------------------------------
// Kernel 3: batched_nms level offset: boxes_off = proposals + ids * (max+1)
// ---------------------------------------------------------------------------
__global__ void pl_offset_kernel(const float* __restrict__ proposals,
                                 const int* __restrict__ ids,
                                 const int* __restrict__ maxbits,
                                 float* __restrict__ boxes_off,
                                 int B, int N) {
    const int total = B * N;
    const float4* p4 = reinterpret_cast<const float4*>(proposals);
    float4* o4 = reinterpret_cast<float4*>(boxes_off);
    for (int id = blockIdx.x * blockDim.x + threadIdx.x; id < total;
         id += gridDim.x * blockDim.x) {
        const int b = id / N;
        const int j = id - b * N;
        const float maxc = __int_as_float(maxbits[b]) + 1.0f;
        const float off = (float)ids[j] * maxc;
        float4 p = p4[id];
        p.x += off; p.y += off; p.z += off; p.w += off;
        o4[id] = p;
    }
}

// ---------------------------------------------------------------------------
// Kernel 4: greedy NMS, one workgroup (32 waves) per image.
// Scores live in LDS (256KB of the 320KB WGP LDS), bulk-loaded by the Tensor
// Data Mover. Each thread caches its 64-score chunk's (max,argmax) in
// registers and rescans only when the chunk actually changed, so the per-step
// argmax is a shuffle tree, not a 256KB LDS sweep. Boxes stream from L2.
// ---------------------------------------------------------------------------
__global__ void __launch_bounds__(NMS_BLOCK)
pl_nms_kernel(const float* __restrict__ scores,
              const float* __restrict__ proposals,
              const float* __restrict__ boxes_off,
              float* __restrict__ out,
              int N) {
    extern __shared__ float smem[];
    float* s_scores = smem;                       // N floats
    float* s_wval   = smem + N;                   // 32
    int*   s_widx   = (int*)(s_wval + 32);        // 32
    float* s_bcast  = (float*)(s_widx + 32);      // [0..3]=top box_off, [4]=valid
    int*   s_bcast_i = (int*)s_bcast;             // [5] = winner index

    const int b    = blockIdx.x;
    const int tid  = threadIdx.x;
    const int lane = tid & 31;
    const int wid  = tid >> 5;
    const int nwaves = blockDim.x >> 5;
    const int C    = N / blockDim.x;              // 64 scores/boxes per thread
    const int base = tid * C;

    const float4* gb4 = reinterpret_cast<const float4*>(boxes_off) + (size_t)b * N;
    const float4* pp4 = reinterpret_cast<const float4*>(proposals) + (size_t)b * N;

    // ---- Stage scores[b, :] into LDS via the Tensor Data Mover -------------
#if defined(HAVE_TDM)
    if (tid == 0) {
        const uint64_t ga  = (uint64_t)(uintptr_t)(scores + (size_t)b * N);
        const uint32_t lds = (uint32_t)(uintptr_t)s_scores;   // low 32b = LDS offset
        const uint32_t W = 1024u;                  // tile/tensor dim0 (elements)
        const uint32_t H = (uint32_t)N / W;        // tile/tensor dim1

        // D# group 0: count=1 | lds_addr | global_addr | type=2
        v4u g0;
        g0[0] = 1u;
        g0[1] = lds;
        g0[2] = (uint32_t)(ga & 0xffffffffu);
        g0[3] = (uint32_t)((ga >> 32) & 0x01ffffffu) | (2u << 30);

        // D# group 1: data_size=4B, tensor 2-D W x H, tile W x H, stride0 = W
        uint32_t d[8] = {0, 0, 0, 0, 0, 0, 0, 0};
        d[0] = (2u << 16);                               // data_size = 4 bytes
        d[1] = (W & 0xffffu) << 16;                      // tensor_dim0[15:0]
        d[2] = (W >> 16) | ((H & 0xffffu) << 16);        // tensor_dim0[31:16]|tensor_dim1[15:0]
        d[3] = (H >> 16) | ((W & 0xffffu) << 16);        // tensor_dim1[31:16]|tile_dim0
        d[4] = (H & 0xffffu);                            // tile_dim1 (tile_dim2=0)
        d[5] = W;                                        // tensor_dim0_stride[31:0]
        d[6] = 0;
        d[7] = 0;
        v8i g1;
        for (int k = 0; k < 8; ++k) g1[k] = (int)d[k];
        v4i z4 = {0, 0, 0, 0};
#if defined(__clang_major__) && (__clang_major__ >= 23)
        v8i z8 = {0, 0, 0, 0, 0, 0, 0, 0};
        __builtin_amdgcn_tensor_load_to_lds(g0, g1, z4, z4, z8, 0);
#else
        __builtin_amdgcn_tensor_load_to_lds(g0, g1, z4, z4, 0);
#endif
    }
#if defined(HAVE_TCNT)
    __builtin_amdgcn_s_wait_tensorcnt(0);
#else
    asm volatile("s_wait_tensorcnt 0" ::: "memory");
#endif
#else
    for (int j = tid; j < N; j += blockDim.x)
        s_scores[j] = scores[(size_t)b * N + j];
#endif
    __syncthreads();

    // ---- 1000 greedy selection steps --------------------------------------
    float cv = NEG_SCORE * 2.0f;   // cached chunk max
    int   ci = base;               // cached chunk argmax
    bool  dirty = true;

    for (int t = 0; t < TOPN; ++t) {
        // refresh cached chunk max only if the chunk changed last step
        if (dirty) {
            cv = NEG_SCORE * 2.0f;
            ci = base;
            for (int k = 0; k < C; k += 4) {
                const float4 q = *reinterpret_cast<const float4*>(&s_scores[base + k]);
                if (q.x > cv) { cv = q.x; ci = base + k; }
                if (q.y > cv) { cv = q.y; ci = base + k + 1; }
                if (q.z > cv) { cv = q.z; ci = base + k + 2; }
                if (q.w > cv) { cv = q.w; ci = base + k + 3; }
            }
            dirty = false;
        }

        // wave32 shuffle argmax (first-index tiebreak like jnp.argmax)
        float bv = cv;
        int   bi = ci;
        for (int off = 16; off > 0; off >>= 1) {
            const float ov = __shfl_down(bv, off);
            const int   oi = __shfl_down(bi, off);
            if (ov > bv || (ov == bv && oi < bi)) { bv = ov; bi = oi; }
        }
        if (lane == 0) { s_wval[wid] = bv; s_widx[wid] = bi; }
        __syncthreads();

        if (wid == 0) {
            float v = (lane < nwaves) ? s_wval[lane] : NEG_SCORE * 2.0f;
            int   i = (lane < nwaves) ? s_widx[lane] : 0;
            for (int off = 16; off > 0; off >>= 1) {
                const float ov = __shfl_down(v, off);
                const int   oi = __shfl_down(i, off);
                if (ov > v || (ov == v && oi < i)) { v = ov; i = oi; }
            }
            if (lane == 0) {
                const bool valid = v > VALID_THRESH;
                const float4 gb = gb4[i];
                const float4 pp = pp4[i];
                s_bcast[0] = gb.x; s_bcast[1] = gb.y;
                s_bcast[2] = gb.z; s_bcast[3] = gb.w;
                s_bcast[4] = valid ? 1.0f : 0.0f;
                s_bcast_i[5] = i;
                s_scores[i] = NEG_SCORE;                 // s.at[i].set(NEG)
                float* orow = out + ((size_t)b * TOPN + t) * 5;
                orow[0] = (float)b;
                orow[1] = valid ? pp.x : 0.0f;
                orow[2] = valid ? pp.y : 0.0f;
                orow[3] = valid ? pp.z : 0.0f;
                orow[4] = valid ? pp.w : 0.0f;
            }
        }
        __syncthreads();

        // winner landed in my chunk -> its score became NEG -> rescan next step
        const int wini = s_bcast_i[5];
        if (wini >= base && wini < base + C) dirty = true;

        // IoU suppression of this thread's 64 boxes against the winner
        const float tx1 = s_bcast[0], ty1 = s_bcast[1];
        const float tx2 = s_bcast[2], ty2 = s_bcast[3];
        if (s_bcast[4] != 0.0f) {
            const float a1 = (tx2 - tx1) * (ty2 - ty1);
            for (int k = 0; k < C; ++k) {
                const int j = base + k;
                const float sc = s_scores[j];
                __builtin_prefetch(&gb4[j + 8], 0, 1);   // global_prefetch
                if (sc > VALID_THRESH) {                 // skip already-dead
                    const float4 bb = gb4[j];
                    const float xx1 = fmaxf(tx1, bb.x);
                    const float yy1 = fmaxf(ty1, bb.y);
                    const float xx2 = fminf(tx2, bb.z);
                    const float yy2 = fminf(ty2, bb.w);
                    const float inter = fmaxf(xx2 - xx1, 0.0f) * fmaxf(yy2 - yy1, 0.0f);
                    const float a2 = (bb.z - bb.x) * (bb.w - bb.y);
                    const float iou = inter / fmaxf(a1 + a2 - inter, 1e-6f);
                    if (iou > NMS_THRESH_C) {
                        s_scores[j] = NEG_SCORE;
                        dirty = true;                    // my chunk changed
                    }
                }
            }
        }
        __syncthreads();
    }
}

// ---------------------------------------------------------------------------
extern "C" void kernel_launch(void* const* d_in, const int* in_sizes, int n_in,
                              void* d_out, int out_size, void* d_ws, size_t ws_size,
                              hipStream_t stream) {
    const float* scores  = (const float*)d_in[0];   // [B,N]
    const float* deltas  = (const float*)d_in[1];   // [B,N,4]
    const float* anchors = (const float*)d_in[2];   // [N,4]
    const float* im_info = (const float*)d_in[3];   // [B,3]
    const int*   ids     = (const int*)d_in[4];     // [N]
    float* out = (float*)d_out;                     // [B,TOPN,5]

    const int N = in_sizes[2] / 4;
    const int B = (N > 0) ? in_sizes[0] / N : 0;
    if (N <= 0 || B <= 0) return;

    // workspace layout
    float* proposals = (float*)d_ws;                          // B*N*4 floats
    float* boxes_off = proposals + (size_t)B * N * 4;         // B*N*4 floats
    int*   maxbits   = (int*)(boxes_off + (size_t)B * N * 4); // B ints

    pl_init_kernel<<<1, 256, 0, stream>>>(maxbits, B);

    const int total = B * N;
    int blocks = (total + 255) / 256;
    if (blocks > 4096) blocks = 4096;
    pl_decode_kernel<<<blocks, 256, 0, stream>>>(deltas, anchors, im_info,
                                                 proposals, maxbits, B, N);
    pl_offset_kernel<<<blocks, 256, 0, stream>>>(proposals, ids, maxbits,
                                                 boxes_off, B, N);

    const size_t smem = (size_t)N * sizeof(float) + (32 + 32 + 16) * sizeof(float);
    pl_nms_kernel<<<B, NMS_BLOCK, smem, stream>>>(scores, proposals, boxes_off,
                                                  out, N);
}